// GCN_46377056862921
// MI455X (gfx1250) — compile-verified
//
#include <hip/hip_runtime.h>

typedef __attribute__((ext_vector_type(2))) float v2f;
typedef __attribute__((ext_vector_type(8))) float v8f;

// ---------------- degree / norm precompute ----------------

__global__ void set_deg_kernel(float* deg, int n) {
    int i = blockIdx.x * blockDim.x + threadIdx.x;
    if (i < n) deg[i] = 1.0f;   // self-loop weight
}

__global__ void accum_deg_kernel(float* deg, const int* __restrict__ dst,
                                 const float* __restrict__ w, int e) {
    int i = blockIdx.x * blockDim.x + threadIdx.x;
    if (i < e) unsafeAtomicAdd(&deg[dst[i]], w[i]);
}

__global__ void dinv_kernel(float* deg, int n) {
    int i = blockIdx.x * blockDim.x + threadIdx.x;
    if (i < n) deg[i] = rsqrtf(deg[i]);   // deg >= 1 always (self loop)
}

__global__ void norm_kernel(const float* __restrict__ dinv, const int* __restrict__ src,
                            const int* __restrict__ dst, const float* __restrict__ w,
                            float* __restrict__ nrm, int e) {
    int i = blockIdx.x * blockDim.x + threadIdx.x;
    if (i < e) nrm[i] = dinv[src[i]] * w[i] * dinv[dst[i]];
}

// ---------------- dense transform via fp32 WMMA ----------------
// One wave computes a 16-row block across all COLS output columns.
// W is staged through LDS in a k-pair-interleaved layout:
//   sW[kp][col][2] = { W[2*kp][col], W[2*kp+1][col] }
// so each WMMA B fragment is a single aligned ds_load_b64 (no repack moves).

template <int COLS>
__global__ __launch_bounds__(256) void gemm_wmma_kernel(
    const float* __restrict__ X, const float* __restrict__ W,
    float* __restrict__ OUT, int nRows) {
    constexpr int K  = 128;
    constexpr int NT = COLS / 16;
    constexpr int CSH = (COLS == 128) ? 7 : 6;

    __shared__ float sW[(K / 2) * COLS * 2];
    for (int i = threadIdx.x; i < K * COLS; i += 256) {
        const int row = i >> CSH;
        const int col = i & (COLS - 1);
        sW[(row >> 1) * (COLS * 2) + col * 2 + (row & 1)] = W[i];
    }
    __syncthreads();

    const int wave = threadIdx.x >> 5;
    const int lane = threadIdx.x & 31;
    const int rb   = blockIdx.x * 8 + wave;       // 16-row block index (wave uniform)
    if (rb * 16 >= nRows) return;

    const int l15  = lane & 15;
    const int hsel = lane >> 4;                   // 0 for lanes 0-15, 1 for 16-31

    // A fragment: row = rb*16 + l15, elements (k + 2*hsel), (k + 2*hsel + 1)
    const float* aBase = X + (size_t)(rb * 16 + l15) * K + 2 * hsel;
    // B fragment: pair index kp = k/2 + hsel, column = c*16 + l15
    const float* bCol  = &sW[hsel * (COLS * 2) + l15 * 2];

    v8f acc[NT] = {};

    for (int k = 0; k < K; k += 4) {
        const v2f A = *(const v2f*)(aBase + k);
        const float* bp = bCol + (k >> 1) * (COLS * 2);
#pragma unroll
        for (int c = 0; c < NT; ++c) {
            const v2f B = *(const v2f*)(bp + c * 32);
            acc[c] = __builtin_amdgcn_wmma_f32_16x16x4_f32(
                false, A, false, B, (short)0, acc[c], false, false);
        }
    }

    // D layout: VGPR r, lanes 0-15 -> M=r, lanes 16-31 -> M=r+8; N = l15
    const int rowBase = rb * 16 + (hsel << 3);
#pragma unroll
    for (int c = 0; c < NT; ++c) {
#pragma unroll
        for (int r = 0; r < 8; ++r)
            OUT[(size_t)(rowBase + r) * COLS + c * 16 + l15] = acc[c][r];
    }
}

// ---------------- aggregation ----------------

// out[n][f] = h[n][f] * dinv[n]^2 + b[f]   (self-loop contribution + bias)
template <int COLS>
__global__ void init_out_kernel(const float* __restrict__ h, const float* __restrict__ dinv,
                                const float* __restrict__ b, float* __restrict__ out,
                                long long total) {
    constexpr int SHIFT = (COLS == 128) ? 7 : 6;
    long long idx = (long long)blockIdx.x * blockDim.x + threadIdx.x;
    if (idx >= total) return;
    int node = (int)(idx >> SHIFT);
    int f    = (int)(idx & (COLS - 1));
    float dv = dinv[node];
    out[idx] = h[idx] * dv * dv + b[f];
}

// wave-per-edge atomic scatter: out[dst] += h[src] * norm
template <int COLS>
__global__ __launch_bounds__(256) void scatter_kernel(
    const float* __restrict__ h, const int* __restrict__ src,
    const int* __restrict__ dst, const float* __restrict__ nrm,
    float* out, int e) {
    const long long gid = (long long)blockIdx.x * blockDim.x + threadIdx.x;
    const long long eid = gid >> 5;
    const int lane = (int)(gid & 31);
    if (eid >= (long long)e) return;
    const int s = src[eid];
    const int d = dst[eid];
    const float nm = nrm[eid];
    if (COLS == 128) {
        const float4 v = *(const float4*)&h[(size_t)s * 128 + lane * 4];
        float* o = &out[(size_t)d * 128 + lane * 4];
        unsafeAtomicAdd(o + 0, v.x * nm);
        unsafeAtomicAdd(o + 1, v.y * nm);
        unsafeAtomicAdd(o + 2, v.z * nm);
        unsafeAtomicAdd(o + 3, v.w * nm);
    } else {
        const float2 v = *(const float2*)&h[(size_t)s * 64 + lane * 2];
        float* o = &out[(size_t)d * 64 + lane * 2];
        unsafeAtomicAdd(o + 0, v.x * nm);
        unsafeAtomicAdd(o + 1, v.y * nm);
    }
}

__global__ void relu_kernel(const float* __restrict__ in, float* __restrict__ out,
                            long long total) {
    long long i = (long long)blockIdx.x * blockDim.x + threadIdx.x;
    if (i < total) out[i] = fmaxf(in[i], 0.0f);
}

// ---------------- driver ----------------

extern "C" void kernel_launch(void* const* d_in, const int* in_sizes, int n_in,
                              void* d_out, int out_size, void* d_ws, size_t ws_size,
                              hipStream_t stream) {
    const float* x   = (const float*)d_in[0];
    const int*  eidx = (const int*)d_in[1];
    const float* ew  = (const float*)d_in[2];
    const float* W1  = (const float*)d_in[3];
    const float* b1  = (const float*)d_in[4];
    const float* W2  = (const float*)d_in[5];
    const float* b2  = (const float*)d_in[6];

    const int N = in_sizes[0] / 128;
    const int E = in_sizes[2];
    const int* src = eidx;
    const int* dst = eidx + E;

    // workspace layout (floats): dinv[N] | norm[E] | h1[N*128] | acc[N*128]
    float* dinv = (float*)d_ws;
    float* nrm  = dinv + N;
    float* h1   = nrm + E;
    float* acc  = h1 + (size_t)N * 128;
    float* out  = (float*)d_out;

    const int T = 256;
    const long long nf1 = (long long)N * 128;
    const long long nf2 = (long long)N * 64;
    const int gemmBlocks = ((N + 15) / 16 + 7) / 8;

    // degree / normalization (shared across both layers)
    set_deg_kernel<<<(N + T - 1) / T, T, 0, stream>>>(dinv, N);
    accum_deg_kernel<<<(E + T - 1) / T, T, 0, stream>>>(dinv, dst, ew, E);
    dinv_kernel<<<(N + T - 1) / T, T, 0, stream>>>(dinv, N);
    norm_kernel<<<(E + T - 1) / T, T, 0, stream>>>(dinv, src, dst, ew, nrm, E);

    // layer 1: h1 = x @ W1 ; acc = aggregate(h1) + b1 ; h1 = relu(acc)
    gemm_wmma_kernel<128><<<gemmBlocks, T, 0, stream>>>(x, W1, h1, N);
    init_out_kernel<128><<<(int)((nf1 + T - 1) / T), T, 0, stream>>>(h1, dinv, b1, acc, nf1);
    scatter_kernel<128><<<(int)(((long long)E * 32 + T - 1) / T), T, 0, stream>>>(
        h1, src, dst, nrm, acc, E);
    relu_kernel<<<(int)((nf1 + T - 1) / T), T, 0, stream>>>(acc, h1, nf1);

    // layer 2: g = h1 @ W2 (reuse acc buffer) ; out = aggregate(g) + b2
    gemm_wmma_kernel<64><<<gemmBlocks, T, 0, stream>>>(h1, W2, acc, N);
    init_out_kernel<64><<<(int)((nf2 + T - 1) / T), T, 0, stream>>>(acc, dinv, b2, out, nf2);
    scatter_kernel<64><<<(int)(((long long)E * 32 + T - 1) / T), T, 0, stream>>>(
        acc, src, dst, nrm, out, E);
}